// DisentangledAttention_37288906064026
// MI455X (gfx1250) — compile-verified
//
#include <hip/hip_runtime.h>
#include <hip/hip_bf16.h>
#include <stdint.h>

// ---------------------------------------------------------------------------
// DeBERTa disentangled attention for MI455X (gfx1250), wave32 + f32 WMMA.
//
// - bd/bc re-expressed as dense NT GEMMs over the 767 distinct relative
//   positions (pos_emb[i,j] depends only on j-i).
// - Big K=512 GEMMs use 4-wave workgroups with double-buffered LDS tiles
//   filled by GLOBAL_LOAD_ASYNC_TO_LDS_B128 (ASYNCcnt / s_wait_asynccnt),
//   16 WMMAs per wave per 16-deep k-chunk.
// ---------------------------------------------------------------------------

typedef __attribute__((ext_vector_type(2))) float v2f;
typedef __attribute__((ext_vector_type(8))) float v8f;

static constexpr int BATCH  = 8;
static constexpr int SEQ    = 384;
static constexpr int EMB    = 512;
static constexpr int NHEAD  = 8;
static constexpr int HDIM   = 64;
static constexpr int NREL   = 767;          // distinct values of j-i+383
static constexpr int RPAD   = 768;          // padded to multiple of 16
static constexpr int POSOFF = 128;          // pos_table row = rel + 128
static constexpr int MROWS  = BATCH * SEQ;  // 3072

// f32 WMMA 16x16x4 fragment layout (wave32):
//   A 16x4 : lane l (l<16): m=l, holds K=0,1 ; lane l>=16: m=l-16, K=2,3
//   B 4x16 : lane l (l<16): n=l, holds K=0,1 ; lane l>=16: n=l-16, K=2,3
//   C 16x16: vgpr r, lane l: m = r + (l<16 ? 0 : 8), n = l & 15
#define WMMA_F32(acc, a, b) \
  __builtin_amdgcn_wmma_f32_16x16x4_f32(false, (a), false, (b), (short)0, (acc), false, false)

// Async global -> LDS 16B copy (gfx1250). ldsoff is a workgroup-relative LDS
// byte offset (low 32 bits of the generic shared pointer); ga is the global
// address. Tracked by ASYNCcnt.
__device__ __forceinline__ void async_copy_b128(uint32_t ldsoff, uint64_t ga) {
  asm volatile("global_load_async_to_lds_b128 %0, %1, off"
               :: "v"(ldsoff), "v"(ga) : "memory");
}
__device__ __forceinline__ void wait_async_le4() {
  asm volatile("s_wait_asynccnt 0x4" ::: "memory");
}
__device__ __forceinline__ void wait_async_zero() {
  asm volatile("s_wait_asynccnt 0x0" ::: "memory");
}

// ---------------------------------------------------------------------------
// Staged NN GEMM: C[M x N] = A[M x K] @ B[K x N] + bias.
// Block = 128 threads (4 waves) -> 64x64 C block; wave w owns rows
// [m0+16w, m0+16w+16) x 64 cols (4 accumulators, A frag reused 4x).
// k-chunks of 16 double-buffered in LDS via async b128 loads.
// rowClamp: A row index clamped to rowClamp-1 (767-row pos_table gather
// padded to 768 GEMM rows). K multiple of 16; M, N multiples of 64.
// ---------------------------------------------------------------------------
__global__ void gemm_nn_bias(const float* __restrict__ A,
                             const float* __restrict__ Bm,
                             const float* __restrict__ bias,
                             float* __restrict__ C,
                             int K, int lda, int ldb, int ldc, int rowClamp) {
  __shared__ __align__(16) float lA[2][64 * 16];   // [row][k]  (4 KB each)
  __shared__ __align__(16) float lB[2][16 * 64];   // [k][col]  (4 KB each)

  const int t    = threadIdx.x;        // 0..127
  const int w    = t >> 5;             // wave 0..3
  const int lane = t & 31;
  const int mr   = lane & 15;
  const int kb   = (lane >> 4) << 1;   // 0 or 2
  const int mtop = (lane >> 4) << 3;   // 0 or 8
  const int m0   = blockIdx.y << 6;
  const int n0   = blockIdx.x << 6;

  // async-load slot mapping (each thread stages 2 A slots + 2 B slots / chunk)
  const int sA0 = t, sA1 = t + 128;              // float4 slots 0..255
  const int rowA0 = sA0 >> 2, kqA0 = sA0 & 3;
  const int rowA1 = sA1 >> 2, kqA1 = sA1 & 3;
  const int krB0 = sA0 >> 4, cqB0 = sA0 & 15;
  const int krB1 = sA1 >> 4, cqB1 = sA1 & 15;

  int ar0 = m0 + rowA0; if (ar0 > rowClamp - 1) ar0 = rowClamp - 1;
  int ar1 = m0 + rowA1; if (ar1 > rowClamp - 1) ar1 = rowClamp - 1;

  const int nchunk = K >> 4;

  auto issue = [&](int c, int buf) {
    const int k0 = c << 4;
    async_copy_b128((uint32_t)(uintptr_t)&lA[buf][sA0 * 4],
                    (uint64_t)(uintptr_t)(A + (size_t)ar0 * lda + k0 + kqA0 * 4));
    async_copy_b128((uint32_t)(uintptr_t)&lA[buf][sA1 * 4],
                    (uint64_t)(uintptr_t)(A + (size_t)ar1 * lda + k0 + kqA1 * 4));
    async_copy_b128((uint32_t)(uintptr_t)&lB[buf][sA0 * 4],
                    (uint64_t)(uintptr_t)(Bm + (size_t)(k0 + krB0) * ldb + n0 + cqB0 * 4));
    async_copy_b128((uint32_t)(uintptr_t)&lB[buf][sA1 * 4],
                    (uint64_t)(uintptr_t)(Bm + (size_t)(k0 + krB1) * ldb + n0 + cqB1 * 4));
  };

  v8f acc0 = {}, acc1 = {}, acc2 = {}, acc3 = {};

  int buf = 0;
  issue(0, 0);
  for (int c = 0; c < nchunk; ++c) {
    if (c + 1 < nchunk) { issue(c + 1, buf ^ 1); wait_async_le4(); }
    else                { wait_async_zero(); }
    __syncthreads();

    const float* As = &lA[buf][(w * 16 + mr) * 16];   // this wave's A rows
    const float* Bs = &lB[buf][0];
    #pragma unroll
    for (int q = 0; q < 4; ++q) {
      const int kk = q * 4 + kb;
      v2f a;
      a.x = As[kk];
      a.y = As[kk + 1];
      const float* b0 = Bs + kk * 64 + mr;
      const float* b1 = Bs + (kk + 1) * 64 + mr;
      v2f bv;
      bv.x = b0[0];  bv.y = b1[0];  acc0 = WMMA_F32(acc0, a, bv);
      bv.x = b0[16]; bv.y = b1[16]; acc1 = WMMA_F32(acc1, a, bv);
      bv.x = b0[32]; bv.y = b1[32]; acc2 = WMMA_F32(acc2, a, bv);
      bv.x = b0[48]; bv.y = b1[48]; acc3 = WMMA_F32(acc3, a, bv);
    }
    __syncthreads();
    buf ^= 1;
  }

  // store 16x64 strip (+bias)
  const int mbase = m0 + w * 16 + mtop;
  #pragma unroll
  for (int nt = 0; nt < 4; ++nt) {
    const int n = n0 + nt * 16 + mr;
    const float bv = bias ? bias[n] : 0.0f;
    const v8f& acc = (nt == 0) ? acc0 : (nt == 1) ? acc1 : (nt == 2) ? acc2 : acc3;
    #pragma unroll
    for (int r = 0; r < 8; ++r) {
      C[(size_t)(mbase + r) * ldc + n] = acc[r] + bv;
    }
  }
}

// ---------------------------------------------------------------------------
// C[M x N] = A[M x K] @ Bt[N x K]^T   (NT). Used for bdf/bcf: per-head
// q/k rows (K=64 slice of E=512) against per-head kp/qp rows.
// ---------------------------------------------------------------------------
__global__ void gemm_nt(const float* __restrict__ A,
                        const float* __restrict__ Bt,
                        float* __restrict__ C,
                        int K, int lda, int ldb, int ldc) {
  const int lane = threadIdx.x;
  const int mr   = lane & 15;
  const int kb   = (lane >> 4) << 1;
  const int mtop = (lane >> 4) << 3;
  const int m0   = blockIdx.y << 4;
  const int n0   = blockIdx.x << 4;

  const float* Arow = A  + (size_t)(m0 + mr) * lda + kb;
  const float* Brow = Bt + (size_t)(n0 + mr) * ldb + kb;

  v8f acc = {};
  #pragma unroll 4
  for (int kk = 0; kk < K; kk += 4) {
    v2f a, b;
    a.x = Arow[kk];
    a.y = Arow[kk + 1];
    b.x = Brow[kk];
    b.y = Brow[kk + 1];
    acc = WMMA_F32(acc, a, b);
  }

  const int n = n0 + mr;
  #pragma unroll
  for (int r = 0; r < 8; ++r) {
    const int m = m0 + mtop + r;
    C[(size_t)m * ldc + n] = acc[r];
  }
}

// ---------------------------------------------------------------------------
// Pre-softmax scores for head h:
//   s[b,h,i,j] = (q_bh[i].k_bh[j] + bdf[b*S+i, j-i+383] + bcf[b*S+j, j-i+383]) / 8
// ac tile via WMMA NT over the 64-wide head slice; bd/bc are gathers.
// grid: (S/16 j-tiles, S/16 i-tiles, B)
// ---------------------------------------------------------------------------
__global__ void score_kernel(const float* __restrict__ q,
                             const float* __restrict__ kmat,
                             const float* __restrict__ bdf,
                             const float* __restrict__ bcf,
                             float* __restrict__ wts,
                             int h) {
  const int lane = threadIdx.x;
  const int mr   = lane & 15;
  const int kb   = (lane >> 4) << 1;
  const int mtop = (lane >> 4) << 3;
  const int i0   = blockIdx.y << 4;
  const int j0   = blockIdx.x << 4;
  const int b    = blockIdx.z;

  const float* Arow = q    + (size_t)(b * SEQ + i0 + mr) * EMB + h * HDIM + kb;
  const float* Brow = kmat + (size_t)(b * SEQ + j0 + mr) * EMB + h * HDIM + kb;

  v8f acc = {};
  #pragma unroll
  for (int kk = 0; kk < HDIM; kk += 4) {
    v2f a, bb;
    a.x  = Arow[kk];
    a.y  = Arow[kk + 1];
    bb.x = Brow[kk];
    bb.y = Brow[kk + 1];
    acc = WMMA_F32(acc, a, bb);
  }

  const int j = j0 + mr;
  #pragma unroll
  for (int r = 0; r < 8; ++r) {
    const int i   = i0 + mtop + r;
    const int rel = j - i + (SEQ - 1);   // in [0, 766]
    const float s = (acc[r]
                     + bdf[(size_t)(b * SEQ + i) * RPAD + rel]
                     + bcf[(size_t)(b * SEQ + j) * RPAD + rel]) * 0.125f;
    wts[((size_t)((b * NHEAD + h) * SEQ + i)) * SEQ + j] = s;
  }
}

// ---------------------------------------------------------------------------
// Row softmax in place over the weights region of d_out. One block per row.
// ---------------------------------------------------------------------------
__global__ void softmax_kernel(float* __restrict__ w, int cols) {
  __shared__ float red[128];
  const int t = threadIdx.x;
  float* p = w + (size_t)blockIdx.x * cols;

  float mx = -3.0e38f;
  for (int j = t; j < cols; j += 128) mx = fmaxf(mx, p[j]);
  red[t] = mx; __syncthreads();
  for (int s = 64; s > 0; s >>= 1) {
    if (t < s) red[t] = fmaxf(red[t], red[t + s]);
    __syncthreads();
  }
  mx = red[0]; __syncthreads();

  float sum = 0.0f;
  for (int j = t; j < cols; j += 128) {
    const float e = expf(p[j] - mx);
    p[j] = e;
    sum += e;
  }
  red[t] = sum; __syncthreads();
  for (int s = 64; s > 0; s >>= 1) {
    if (t < s) red[t] += red[t + s];
    __syncthreads();
  }
  const float inv = 1.0f / red[0];
  for (int j = t; j < cols; j += 128) p[j] *= inv;
}

// ---------------------------------------------------------------------------
// o[b, i, h*64+d] = sum_j weights[b,h,i,j] * v[b*S+j, h*64+d]   (NN per b,h)
// grid: (HDIM/16, SEQ/16, B*NHEAD)
// ---------------------------------------------------------------------------
__global__ void av_gemm(const float* __restrict__ wts,
                        const float* __restrict__ vmat,
                        float* __restrict__ o) {
  const int lane = threadIdx.x;
  const int mr   = lane & 15;
  const int kb   = (lane >> 4) << 1;
  const int mtop = (lane >> 4) << 3;
  const int n0   = blockIdx.x << 4;
  const int i0   = blockIdx.y << 4;
  const int bh   = blockIdx.z;
  const int b    = bh >> 3;
  const int h    = bh & 7;

  const float* Arow = wts  + ((size_t)bh * SEQ + i0 + mr) * SEQ + kb;
  const float* Bp   = vmat + (size_t)(b * SEQ + kb) * EMB + h * HDIM + n0 + mr;

  v8f acc = {};
  #pragma unroll 4
  for (int kk = 0; kk < SEQ; kk += 4) {
    v2f a, bb;
    a.x  = Arow[kk];
    a.y  = Arow[kk + 1];
    bb.x = Bp[(size_t)kk * EMB];
    bb.y = Bp[(size_t)(kk + 1) * EMB];
    acc = WMMA_F32(acc, a, bb);
  }

  #pragma unroll
  for (int r = 0; r < 8; ++r) {
    const int i = i0 + mtop + r;
    o[(size_t)(b * SEQ + i) * EMB + h * HDIM + n0 + mr] = acc[r];
  }
}

// ---------------------------------------------------------------------------
extern "C" void kernel_launch(void* const* d_in, const int* in_sizes, int n_in,
                              void* d_out, int out_size, void* d_ws, size_t ws_size,
                              hipStream_t stream) {
  const float* x    = (const float*)d_in[0];
  const float* qW   = (const float*)d_in[1];
  const float* qb   = (const float*)d_in[2];
  const float* kW   = (const float*)d_in[3];
  const float* kb_  = (const float*)d_in[4];
  const float* vW   = (const float*)d_in[5];
  const float* vb   = (const float*)d_in[6];
  const float* qpW  = (const float*)d_in[7];
  const float* qpb  = (const float*)d_in[8];
  const float* kpW  = (const float*)d_in[9];
  const float* kpb  = (const float*)d_in[10];
  const float* pos  = (const float*)d_in[11];
  const float* fcW  = (const float*)d_in[12];
  const float* fcb  = (const float*)d_in[13];

  float* out = (float*)d_out;                       // [B, S, E]
  float* wts = out + (size_t)BATCH * SEQ * EMB;     // [B, H, S, S]

  // Workspace layout (~45 MiB of floats)
  float* ws   = (float*)d_ws;
  float* wq   = ws;                                 // [3072, 512]
  float* wk   = wq  + (size_t)MROWS * EMB;          // [3072, 512]
  float* wv   = wk  + (size_t)MROWS * EMB;          // [3072, 512]
  float* wo   = wv  + (size_t)MROWS * EMB;          // [3072, 512]
  float* wqp  = wo  + (size_t)MROWS * EMB;          // [768, 512]
  float* wkp  = wqp + (size_t)RPAD  * EMB;          // [768, 512]
  float* wbdf = wkp + (size_t)RPAD  * EMB;          // [3072, 768] per-head
  float* wbcf = wbdf + (size_t)MROWS * RPAD;        // [3072, 768] per-head

  const dim3 wave(32);
  const dim3 blk4(128);

  // 1) content projections: q/k/v = x @ W + b    ([3072x512] x [512x512])
  {
    const dim3 g(EMB / 64, MROWS / 64);
    gemm_nn_bias<<<g, blk4, 0, stream>>>(x, qW, qb,  wq, EMB, EMB, EMB, EMB, MROWS);
    gemm_nn_bias<<<g, blk4, 0, stream>>>(x, kW, kb_, wk, EMB, EMB, EMB, EMB, MROWS);
    gemm_nn_bias<<<g, blk4, 0, stream>>>(x, vW, vb,  wv, EMB, EMB, EMB, EMB, MROWS);
  }

  // 2) position projections over the 767 distinct relative rows
  //    (pos_table rows 128..894), padded to 768 GEMM rows via row clamp.
  {
    const dim3 g(EMB / 64, RPAD / 64);
    gemm_nn_bias<<<g, blk4, 0, stream>>>(pos + (size_t)POSOFF * EMB, qpW, qpb,
                                         wqp, EMB, EMB, EMB, EMB, NREL);
    gemm_nn_bias<<<g, blk4, 0, stream>>>(pos + (size_t)POSOFF * EMB, kpW, kpb,
                                         wkp, EMB, EMB, EMB, EMB, NREL);
  }

  // 3) per head: bdf = Q_h @ KP_h^T, bcf = K_h @ QP_h^T, then scores.
  for (int h = 0; h < NHEAD; ++h) {
    const int co = h * HDIM;
    const dim3 gnt(RPAD / 16, MROWS / 16);
    gemm_nt<<<gnt, wave, 0, stream>>>(wq + co, wkp + co, wbdf, HDIM, EMB, EMB, RPAD);
    gemm_nt<<<gnt, wave, 0, stream>>>(wk + co, wqp + co, wbcf, HDIM, EMB, EMB, RPAD);
    const dim3 gsc(SEQ / 16, SEQ / 16, BATCH);
    score_kernel<<<gsc, wave, 0, stream>>>(wq, wk, wbdf, wbcf, wts, h);
  }

  // 4) softmax over j, in place in d_out's weights region
  softmax_kernel<<<dim3(BATCH * NHEAD * SEQ), dim3(128), 0, stream>>>(wts, SEQ);

  // 5) attention output: o = weights @ v
  av_gemm<<<dim3(HDIM / 16, SEQ / 16, BATCH * NHEAD), wave, 0, stream>>>(wts, wv, wo);

  // 6) final projection: out = o @ fcW + fcb
  gemm_nn_bias<<<dim3(EMB / 64, MROWS / 64), blk4, 0, stream>>>(
      wo, fcW, fcb, out, EMB, EMB, EMB, EMB, MROWS);
}